// ParallelFreqAwareEmbeddingBagTablewise_33131377721825
// MI455X (gfx1250) — compile-verified
//
#include <hip/hip_runtime.h>

// EmbeddingBag (mode='mean'), tablewise: T=8 tables, D=128, bags = T*B.
// weights [T,R,D] fp32 contiguous; indices are GLOBAL row ids, so the gather
// address is simply weights + idx*D.
//
// Memory-bound (AI ~ 0.25 FLOP/B; floor ~36us at 23.3 TB/s). Strategy:
//   - one wave32 per bag; lane owns 4 floats -> 512B row = 32 x b128 loads
//   - bag id forced wave-uniform via readfirstlane so offsets/indices go
//     through SCALAR loads (s_load, KMcnt) and address math stays in SALU;
//     the vector path carries only the row gathers
//   - 4-deep unroll: 4 independent global_load_b128 gathers in flight
//   - global_prefetch_b8 8 rows ahead to hide random-access HBM latency

#define TBL 8
#define DIM 128   // floats per row; 32 lanes x 4 floats

__global__ __launch_bounds__(256) void embbag_mean_kernel(
    const float* __restrict__ w,
    const int*   __restrict__ indices,
    const int*   __restrict__ offsets,
    float*       __restrict__ out,
    int num_bags, int n_idx, int batch)
{
  const int lane = threadIdx.x & 31;

  // One bag per wave32; value is wave-uniform -> tell the compiler so.
  int bag = blockIdx.x * (blockDim.x >> 5) + (threadIdx.x >> 5);
  bag = __builtin_amdgcn_readfirstlane(bag);
  if (bag >= num_bags) return;

  // Uniform control data: these become scalar (s_load) accesses.
  const int start = offsets[bag];
  const int end   = (bag + 1 < num_bags) ? offsets[bag + 1] : n_idx;
  const int cnt   = end - start;

  const int table = bag / batch;          // t   (scalar division, once/wave)
  const int brow  = bag - table * batch;  // b

  const int laneoff = lane * 4;           // float offset within the row

  float a0x = 0.f, a0y = 0.f, a0z = 0.f, a0w = 0.f;
  float a1x = 0.f, a1y = 0.f, a1z = 0.f, a1w = 0.f;
  float a2x = 0.f, a2y = 0.f, a2z = 0.f, a2w = 0.f;
  float a3x = 0.f, a3y = 0.f, a3z = 0.f, a3w = 0.f;

  const int PF = 8;  // prefetch-ahead distance (rows)

  // Warm the pipe: prefetch the first PF rows of this bag.
  for (int p = start; p < start + PF && p < end; ++p) {
    const float* row = w + (size_t)indices[p] * DIM;
    __builtin_prefetch(row + laneoff, 0, 1);
  }

  int j = start;
  for (; j + 3 < end; j += 4) {
    // Wave-uniform index fetches -> scalar loads.
    const int i0 = indices[j];
    const int i1 = indices[j + 1];
    const int i2 = indices[j + 2];
    const int i3 = indices[j + 3];

    if (j + PF + 3 < end) {
      __builtin_prefetch(w + (size_t)indices[j + PF]     * DIM + laneoff, 0, 1);
      __builtin_prefetch(w + (size_t)indices[j + PF + 1] * DIM + laneoff, 0, 1);
      __builtin_prefetch(w + (size_t)indices[j + PF + 2] * DIM + laneoff, 0, 1);
      __builtin_prefetch(w + (size_t)indices[j + PF + 3] * DIM + laneoff, 0, 1);
    }

    // 4 independent 512B row gathers in flight (global_load_b128 each).
    const float4 v0 = *(const float4*)(w + (size_t)i0 * DIM + laneoff);
    const float4 v1 = *(const float4*)(w + (size_t)i1 * DIM + laneoff);
    const float4 v2 = *(const float4*)(w + (size_t)i2 * DIM + laneoff);
    const float4 v3 = *(const float4*)(w + (size_t)i3 * DIM + laneoff);

    a0x += v0.x; a0y += v0.y; a0z += v0.z; a0w += v0.w;
    a1x += v1.x; a1y += v1.y; a1z += v1.z; a1w += v1.w;
    a2x += v2.x; a2y += v2.y; a2z += v2.z; a2w += v2.w;
    a3x += v3.x; a3y += v3.y; a3z += v3.z; a3w += v3.w;
  }
  for (; j < end; ++j) {
    const int i0 = indices[j];
    const float4 v0 = *(const float4*)(w + (size_t)i0 * DIM + laneoff);
    a0x += v0.x; a0y += v0.y; a0z += v0.z; a0w += v0.w;
  }

  const float scale = 1.0f / (float)(cnt > 1 ? cnt : 1);

  float4 res;
  res.x = ((a0x + a1x) + (a2x + a3x)) * scale;
  res.y = ((a0y + a1y) + (a2y + a3y)) * scale;
  res.z = ((a0z + a1z) + (a2z + a3z)) * scale;
  res.w = ((a0w + a1w) + (a2w + a3w)) * scale;

  // out[b, t*D + d]  (shape [B, T*D])
  float* dst = out + (size_t)brow * (TBL * DIM) + (size_t)table * DIM + laneoff;
  *(float4*)dst = res;   // global_store_b128
}

extern "C" void kernel_launch(void* const* d_in, const int* in_sizes, int n_in,
                              void* d_out, int out_size, void* d_ws, size_t ws_size,
                              hipStream_t stream) {
  const float* w       = (const float*)d_in[0];  // [T,R,D] fp32
  const int*   indices = (const int*)  d_in[1];  // [n] global row ids
  const int*   offsets = (const int*)  d_in[2];  // [T*B] bag starts
  float*       out     = (float*)      d_out;    // [B, T*D] fp32

  const int n_idx    = in_sizes[1];
  const int num_bags = in_sizes[2];
  const int batch    = num_bags / TBL;           // B

  const int wavesPerBlock = 8;                   // 256 threads = 8 wave32
  const int blocks = (num_bags + wavesPerBlock - 1) / wavesPerBlock;

  embbag_mean_kernel<<<dim3(blocks), dim3(256), 0, stream>>>(
      w, indices, offsets, out, num_bags, n_idx, batch);
}